// Hessian_49160195670508
// MI455X (gfx1250) — compile-verified
//
#include <hip/hip_runtime.h>
#include <hip/hip_bf16.h>

// MI455X (gfx1250) fused Hessian-response kernel.
// Memory-bound: ~470 MB nominal traffic (recompute strategy), second pass of
// the 201 MB input largely served from the 192 MB L2 -> ~12-14 us floor.
// Blur is executed as banded matmuls on V_WMMA_F32_16X16X4_F32.

typedef __attribute__((ext_vector_type(2))) float v2f;
typedef __attribute__((ext_vector_type(8))) float v8f;

#define HW     1024
#define TILE   64
#define G_N    68            // blurred-g region per side (out + 2*2 halo)
#define GRAY_N 72            // grayscale region (g + 2*2 blur halo)
#define GX_N   66            // first-derivative region (out + 2*1 halo)
#define SG_GRAY 73           // odd LDS strides -> conflict-free column walks
#define SG_T    69
#define SG_G    69
#define SG_X    67

#define G_FLOATS    (G_N * SG_G)        // 4692
#define GRAY_FLOATS (GRAY_N * SG_GRAY)  // 5256
#define T_FLOATS    (GRAY_N * SG_T)     // 4968
#define OFF_G   0
#define OFF_A   (G_FLOATS)              // phase-A scratch base (gray), reused for gx
#define OFF_T   (OFF_A + GRAY_FLOATS)   // horizontal-blur temp
#define OFF_GX  (OFF_A)
#define OFF_GY  (OFF_A + GX_N * SG_X)
#define SMEM_FLOATS (OFF_T + T_FLOATS)  // 14916 floats = 59664 B < 64 KB

// normalized 1D Gaussian (L=5, sig=10); outer(w,w)/sum == separable wn per axis
__device__ __forceinline__ float bandw(int d) {
  float r = 0.0f;
  if (d == 0 || d == 4) r = 0.19800304f;
  else if (d == 1 || d == 3) r = 0.20099548f;
  else if (d == 2) r = 0.20200297f;
  return r;
}

template <bool STATS>
__global__ __launch_bounds__(256) void hessian_tile(
    const float* __restrict__ x, const float* __restrict__ to_gray,
    float* __restrict__ out, unsigned int* __restrict__ smax) {
  __shared__ float smem[SMEM_FLOATS];
  float* sG    = smem + OFF_G;
  float* sGray = smem + OFF_A;
  float* sT    = smem + OFF_T;

  const int tid  = threadIdx.x;
  const int lane = tid & 31;
  const int wave = tid >> 5;
  const int b    = blockIdx.z;
  const int R0   = blockIdx.y * TILE;
  const int C0   = blockIdx.x * TILE;

  const float w0 = to_gray[0], w1 = to_gray[1], w2 = to_gray[2];
  const float* xb = x + (size_t)b * 3u * HW * HW;

  // ---- stage 1: grayscale tile into LDS, zero outside image (SAME pad) ----
  for (int idx = tid; idx < GRAY_N * GRAY_N; idx += 256) {
    int rr = idx / GRAY_N, cc = idx % GRAY_N;
    int gr = R0 - 4 + rr, gc = C0 - 4 + cc;
    float v = 0.0f;
    if ((unsigned)gr < HW && (unsigned)gc < HW) {
      size_t p = (size_t)gr * HW + gc;
      v = w0 * xb[p] + w1 * xb[p + (size_t)HW * HW] + w2 * xb[p + (size_t)2 * HW * HW];
    }
    sGray[rr * SG_GRAY + cc] = v;
  }
  __syncthreads();

  // ---- stage 2: horizontal blur as WMMA: T(16x16) = Gray(16x20) x Band(20x16)
  // A f32 16x4 layout: lane M=lane&15; VGPR0 holds K = (lane<16?0:2), VGPR1 K+1.
  // B 4x16 assumed K-striped the same way; col N = lane&15.
  {
    const int M  = lane & 15;
    const int ks = (lane >> 4) << 1;
    for (int t5 = wave; t5 < 25; t5 += 8) {
      int m0 = (t5 / 5) * 16; if (m0 > GRAY_N - 16) m0 = GRAY_N - 16;   // {0,16,32,48,56}
      int n0 = (t5 % 5) * 16; if (n0 > G_N   - 16) n0 = G_N   - 16;     // {0,16,32,48,52}
      v8f acc = {0.f, 0.f, 0.f, 0.f, 0.f, 0.f, 0.f, 0.f};
#pragma unroll
      for (int s = 0; s < 5; ++s) {
        int k0 = 4 * s + ks;
        v2f a, bm;
        a.x  = sGray[(m0 + M) * SG_GRAY + n0 + k0];
        a.y  = sGray[(m0 + M) * SG_GRAY + n0 + k0 + 1];
        bm.x = bandw(k0 - M);          // B[k][n] = w[k-n], n == lane&15
        bm.y = bandw(k0 + 1 - M);
        acc = __builtin_amdgcn_wmma_f32_16x16x4_f32(false, a, false, bm,
                                                    (short)0, acc, false, false);
      }
#pragma unroll
      for (int v = 0; v < 8; ++v) {
        int m = m0 + v + ((lane >> 4) << 3);
        sT[m * SG_T + n0 + M] = acc[v];       // overlapped tiles: same-value writes
      }
    }
  }
  __syncthreads();

  // ---- stage 3: vertical blur as WMMA: G(16x16) = Band(16x20) x T(20x16) ----
  {
    const int M  = lane & 15;
    const int ks = (lane >> 4) << 1;
    for (int t5 = wave; t5 < 25; t5 += 8) {
      int gi0 = (t5 / 5) * 16; if (gi0 > G_N - 16) gi0 = G_N - 16;  // {0,16,32,48,52}
      int gj0 = (t5 % 5) * 16; if (gj0 > G_N - 16) gj0 = G_N - 16;
      v8f acc = {0.f, 0.f, 0.f, 0.f, 0.f, 0.f, 0.f, 0.f};
#pragma unroll
      for (int s = 0; s < 5; ++s) {
        int k0 = 4 * s + ks;
        v2f a, bm;
        a.x  = bandw(k0 - M);          // A[m][k] = w[k-m], m == lane&15
        a.y  = bandw(k0 + 1 - M);
        bm.x = sT[(gi0 + k0) * SG_T + gj0 + M];
        bm.y = sT[(gi0 + k0 + 1) * SG_T + gj0 + M];
        acc = __builtin_amdgcn_wmma_f32_16x16x4_f32(false, a, false, bm,
                                                    (short)0, acc, false, false);
      }
#pragma unroll
      for (int v = 0; v < 8; ++v) {
        int m = gi0 + v + ((lane >> 4) << 3);
        sG[m * SG_G + gj0 + M] = acc[v];
      }
    }
  }
  __syncthreads();

  // ---- stage 4: first derivatives (torch.gradient semantics) into LDS ----
  float* sGx = smem + OFF_GX;   // aliases dead gray/T scratch (barriered)
  float* sGy = smem + OFF_GY;
  for (int idx = tid; idx < GX_N * GX_N; idx += 256) {
    int i = idx / GX_N, j = idx % GX_N;
    int gi = i + 1, gj = j + 1;
    int r = R0 - 2 + gi, c = C0 - 2 + gj;
    float ctr = sG[gi * SG_G + gj];
    float up = sG[(gi - 1) * SG_G + gj], dn = sG[(gi + 1) * SG_G + gj];
    float lf = sG[gi * SG_G + gj - 1],   rt = sG[gi * SG_G + gj + 1];
    float gx = (r <= 0) ? (dn - ctr) : (r >= HW - 1) ? (ctr - up) : 0.5f * (dn - up);
    float gy = (c <= 0) ? (rt - ctr) : (c >= HW - 1) ? (ctr - lf) : 0.5f * (rt - lf);
    sGx[i * SG_X + j] = gx;
    sGy[i * SG_X + j] = gy;
  }
  __syncthreads();

  // ---- stage 5: Hessian -> eigenvalues -> s (and response on pass 2) ----
  float gamma_inv2 = 0.0f;
  if (!STATS) {
    bool allz = true;
#pragma unroll
    for (int k = 0; k < 16; ++k)
      if (__uint_as_float(smax[k]) != 0.0f) allz = false;
    float gamma = 0.5f * __uint_as_float(smax[b]);
    if (allz) gamma = 1.0f;
    gamma_inv2 = 1.0f / (2.0f * gamma * gamma);
  }
  float lmax = 0.0f;
  for (int idx = tid; idx < TILE * TILE; idx += 256) {
    int i = idx / TILE, j = idx % TILE;
    int r = R0 + i, c = C0 + j;
    float a0 = sGx[(i)     * SG_X + (j + 1)];
    float a1 = sGx[(i + 1) * SG_X + (j + 1)];
    float a2 = sGx[(i + 2) * SG_X + (j + 1)];
    float b0 = sGx[(i + 1) * SG_X + (j)];
    float b2 = sGx[(i + 1) * SG_X + (j + 2)];
    float c0v = sGy[(i + 1) * SG_X + (j)];
    float c1  = sGy[(i + 1) * SG_X + (j + 1)];
    float c2  = sGy[(i + 1) * SG_X + (j + 2)];
    float h00 = (r == 0) ? (a2 - a1) : (r == HW - 1) ? (a1 - a0) : 0.5f * (a2 - a0);
    float h01 = (c == 0) ? (b2 - a1) : (c == HW - 1) ? (a1 - b0) : 0.5f * (b2 - b0);
    float h11 = (c == 0) ? (c2 - c1) : (c == HW - 1) ? (c1 - c0v) : 0.5f * (c2 - c0v);
    float mean = 0.5f * (h00 + h11);
    float hd   = 0.5f * (h00 - h11);
    float disc = sqrtf(hd * hd + h01 * h01);
    float e0 = mean - disc, e1 = mean + disc;
    float s2 = e0 * e0 + e1 * e1;
    float s  = sqrtf(s2);
    if (STATS) {
      lmax = fmaxf(lmax, s);
    } else {
      bool sw = fabsf(e1) < fabsf(e0);
      float lam1 = sw ? e1 : e0;
      float lam2 = fmaxf(sw ? e0 : e1, 1e-10f);
      float rb = fabsf(lam1) / lam2;
      float vals = expf(-2.0f * rb * rb) * (1.0f - expf(-s2 * gamma_inv2));
      float f = fmaxf(0.0f, vals);
      out[((size_t)b * HW + r) * HW + c] = (f <= 0.0f) ? 1.0f : f;
    }
  }
  if (STATS) {
    __syncthreads();
    float* red = smem;        // sG dead here
    red[tid] = lmax;
    __syncthreads();
    for (int off = 128; off > 0; off >>= 1) {
      if (tid < off) red[tid] = fmaxf(red[tid], red[tid + off]);
      __syncthreads();
    }
    // s >= 0, so uint compare == float compare; max is order-independent.
    if (tid == 0) atomicMax(&smax[b], __float_as_uint(red[0]));
  }
}

__global__ void init_ws(unsigned int* p) {
  if (threadIdx.x < 16) p[threadIdx.x] = 0u;
}

extern "C" void kernel_launch(void* const* d_in, const int* in_sizes, int n_in,
                              void* d_out, int out_size, void* d_ws, size_t ws_size,
                              hipStream_t stream) {
  const float* x  = (const float*)d_in[0];
  const float* tg = (const float*)d_in[1];
  float* out = (float*)d_out;
  unsigned int* smax = (unsigned int*)d_ws;   // 16 per-batch max(s) as uint bits

  init_ws<<<1, 32, 0, stream>>>(smax);
  dim3 grid(HW / TILE, HW / TILE, 16);
  hessian_tile<true ><<<grid, 256, 0, stream>>>(x, tg, out, smax);  // stats pass
  hessian_tile<false><<<grid, 256, 0, stream>>>(x, tg, out, smax);  // finalize
}